// NeuralSumProductModel_90838558311073
// MI455X (gfx1250) — compile-verified
//
#include <hip/hip_runtime.h>
#include <stdint.h>

#define N_VARS   8192
#define N_CHKS   4096
#define DVE      3
#define DCE      6
#define N_EDGES  24576        // N_VARS*DVE == N_CHKS*DCE
#define BATCHN   512
#define N_ITERS  5

typedef __attribute__((ext_vector_type(2))) float v2f;
typedef __attribute__((ext_vector_type(8))) float v8f;

// ---------------------------------------------------------------------------
// Build chk->edge table (atomic slot assignment; order fixed by sort kernel).
// ---------------------------------------------------------------------------
__global__ void k_build_chk2edge(const int* __restrict__ chk_index,
                                 int* __restrict__ cnt,
                                 int* __restrict__ chk2edge) {
    int e = blockIdx.x * blockDim.x + threadIdx.x;
    if (e < N_EDGES) {
        int c = chk_index[e];
        int s = atomicAdd(&cnt[c], 1);
        chk2edge[c * DCE + s] = e;
    }
}

// Sort each check's 6 edge ids ascending -> deterministic accumulation order
// (matches index-ordered segment_sum of the reference).
__global__ void k_sort_chk2edge(int* __restrict__ chk2edge) {
    int c = blockIdx.x * blockDim.x + threadIdx.x;
    if (c >= N_CHKS) return;
    int v[DCE];
#pragma unroll
    for (int i = 0; i < DCE; ++i) v[i] = chk2edge[c * DCE + i];
#pragma unroll
    for (int i = 1; i < DCE; ++i) {
        int key = v[i];
        int j = i - 1;
        while (j >= 0 && v[j] > key) { v[j + 1] = v[j]; --j; }
        v[j + 1] = key;
    }
#pragma unroll
    for (int i = 0; i < DCE; ++i) chk2edge[c * DCE + i] = v[i];
}

// ---------------------------------------------------------------------------
// Exact f32 transpose via V_WMMA_F32_16X16X4_F32 with identity B.
// One wave per 16x16 tile:  in[R][C] -> out[C][R]  (R, C multiples of 16).
//
// A-operand layout (ISA 7.12.2, 32-bit A 16x4): chunk j VGPR0 holds
// A[m][4j]   on lanes 0-15 / A[m][4j+2] on lanes 16-31; VGPR1 holds
// A[m][4j+1] / A[m][4j+3].  We place A = T^t, so each VGPR fill is a
// coalesced 64B row segment of `in`.  B_j = rows 4j..4j+3 of I16 =>
// D = T^t in row-striped D layout => coalesced 64B row stores of `out`.
// Exact: each output = 1.0*x + 0*y + 0*y + 0*y under RNE.
// ---------------------------------------------------------------------------
__global__ void __launch_bounds__(512)
k_wmma_transpose(const float* __restrict__ in, float* __restrict__ out,
                 int R, int C) {
    int wave = (blockIdx.x * blockDim.x + threadIdx.x) >> 5;
    int lane = threadIdx.x & 31;
    int tilesX = C >> 4;
    int tx = wave % tilesX;
    int ty = wave / tilesX;
    int r0 = ty << 4, c0 = tx << 4;

    int half = lane >> 4;       // 0: lanes 0-15, 1: lanes 16-31
    int li   = lane & 15;

    v8f acc = {};
#pragma unroll
    for (int j = 0; j < 4; ++j) {
        v2f a, b;
        int ra = r0 + 4 * j + (half ? 2 : 0);
        a.x = in[(size_t)ra       * C + c0 + li];   // A[li][4j + (half?2:0)] = T^t
        a.y = in[(size_t)(ra + 1) * C + c0 + li];   // A[li][4j + (half?3:1)]
        int k0 = 4 * j + (half ? 2 : 0);
        int k1 = 4 * j + (half ? 3 : 1);
        b.x = (li == k0) ? 1.0f : 0.0f;             // B_j[k][n] = delta(n, 4j+k)
        b.y = (li == k1) ? 1.0f : 0.0f;
        acc = __builtin_amdgcn_wmma_f32_16x16x4_f32(
            /*neg_a=*/false, a, /*neg_b=*/false, b,
            /*c_mod=*/(short)0, acc, /*reuse_a=*/false, /*reuse_b=*/false);
    }
#pragma unroll
    for (int r = 0; r < 8; ++r) {
        int m = r + (half ? 8 : 0);                 // D row m = T^t row m
        out[(size_t)(c0 + m) * R + r0 + li] = acc[r];
    }
}

// ---------------------------------------------------------------------------
// msg init: ext==0  =>  msg[e][b] = llrT[var(e)][b]
// ---------------------------------------------------------------------------
__global__ void k_init_msg(const float* __restrict__ llrT,
                           const int* __restrict__ var_index,
                           float* __restrict__ m) {
    long long idx = (long long)blockIdx.x * blockDim.x + threadIdx.x;
    if (idx < (long long)N_EDGES * BATCHN) {
        int e = (int)(idx >> 9);           // BATCHN == 512
        int b = (int)(idx & (BATCHN - 1));
        int v = var_index[e];
        m[idx] = llrT[(size_t)v * BATCHN + b];
    }
}

// ---------------------------------------------------------------------------
// Check-node kernel: one block per check, one thread per batch column.
// Stages the 6 message rows through LDS with CDNA5 async global->LDS loads,
// then computes the log-domain leave-one-out product and writes ext in-place.
// ---------------------------------------------------------------------------
__global__ void __launch_bounds__(BATCHN)
k_check(float* __restrict__ m, const int* __restrict__ chk2edge) {
    __shared__ float smem[DCE * BATCHN];              // 12 KB
    const int c   = blockIdx.x;
    const int tid = threadIdx.x;

    int rows[DCE];
#pragma unroll
    for (int k = 0; k < DCE; ++k) rows[k] = chk2edge[c * DCE + k];

    // Async copy: lane tid pulls element tid of each of the 6 rows into LDS.
#pragma unroll
    for (int k = 0; k < DCE; ++k) {
        const float* gp = m + (size_t)rows[k] * BATCHN + tid;
        unsigned lds_off = (unsigned)(uintptr_t)&smem[k * BATCHN + tid];
        asm volatile("global_load_async_to_lds_b32 %0, %1, off"
                     :: "v"(lds_off), "v"(gp) : "memory");
    }
    asm volatile("s_wait_asynccnt 0" ::: "memory");

    float la[DCE], sg[DCE];
    float csum = 0.0f, sprod = 1.0f;
#pragma unroll
    for (int k = 0; k < DCE; ++k) {
        float mm = smem[k * BATCHN + tid];
        float t  = tanhf(0.5f * mm);
        sg[k] = (t > 0.0f) ? 1.0f : ((t < 0.0f) ? -1.0f : 0.0f);
        la[k] = __logf(fabsf(t) + 1e-12f);
        csum += la[k];
        sprod *= sg[k];
    }
#pragma unroll
    for (int k = 0; k < DCE; ++k) {
        float loo = __expf(csum - la[k]) * (sprod * sg[k]);
        loo = fminf(fmaxf(loo, -0.99999994f), 0.99999994f);   // 1 - 1e-7 (f32)
        float ext = __logf((1.0f + loo) / (1.0f - loo));      // 2*atanh(loo)
        m[(size_t)rows[k] * BATCHN + tid] = ext;
    }
}

// ---------------------------------------------------------------------------
// Variable-node kernel: one block per variable, one thread per batch column.
// Edges of variable v are rows 3v..3v+2 (contiguous). Produces the iteration
// output (var-major) and, unless last iteration, next-round messages in-place.
// ---------------------------------------------------------------------------
__global__ void __launch_bounds__(BATCHN)
k_var(float* __restrict__ m, const float* __restrict__ llrT,
      float* __restrict__ outT, int last) {
    const int v   = blockIdx.x;
    const int tid = threadIdx.x;
    size_t base = (size_t)v * DVE * BATCHN + tid;

    float e0 = m[base];
    float e1 = m[base + BATCHN];
    float e2 = m[base + 2 * BATCHN];
    float L  = llrT[(size_t)v * BATCHN + tid];
    float vsum = e0 + e1 + e2;

    outT[(size_t)v * BATCHN + tid] = vsum + L;

    if (!last) {
        m[base]              = (vsum - e0) + L;
        m[base + BATCHN]     = (vsum - e1) + L;
        m[base + 2 * BATCHN] = (vsum - e2) + L;
    }
}

// ---------------------------------------------------------------------------
extern "C" void kernel_launch(void* const* d_in, const int* in_sizes, int n_in,
                              void* d_out, int out_size, void* d_ws, size_t ws_size,
                              hipStream_t stream) {
    (void)in_sizes; (void)n_in; (void)out_size; (void)ws_size;

    const float* llr       = (const float*)d_in[0];   // [512][8192]
    const int*   var_index = (const int*)  d_in[1];   // [24576]
    const int*   chk_index = (const int*)  d_in[2];   // [24576]
    float*       out       = (float*)d_out;           // [5][512][8192]

    char* ws = (char*)d_ws;
    float* llrT     = (float*)ws;                                   // 16 MB
    float* m        = (float*)(ws + ((size_t)16 << 20));            // 48 MB
    float* outT     = (float*)(ws + ((size_t)64 << 20));            // 16 MB
    int*   chk2edge = (int*)  (ws + ((size_t)80 << 20));            // 96 KB
    int*   cnt      = (int*)  (ws + ((size_t)80 << 20) + 98304);    // 16 KB

    hipMemsetAsync(cnt, 0, N_CHKS * sizeof(int), stream);
    k_build_chk2edge<<<(N_EDGES + 255) / 256, 256, 0, stream>>>(chk_index, cnt, chk2edge);
    k_sort_chk2edge<<<(N_CHKS + 255) / 256, 256, 0, stream>>>(chk2edge);

    // tiles: (R/16)*(C/16), 16 waves (one tile each) per 512-thread block
    // llr [512][8192] -> llrT [8192][512]
    {
        int blocks = ((BATCHN / 16) * (N_VARS / 16)) / 16;   // 1024
        k_wmma_transpose<<<blocks, 512, 0, stream>>>(llr, llrT, BATCHN, N_VARS);
    }

    long long tot = (long long)N_EDGES * BATCHN;
    k_init_msg<<<(int)((tot + 511) / 512), 512, 0, stream>>>(llrT, var_index, m);

    for (int it = 0; it < N_ITERS; ++it) {
        k_check<<<N_CHKS, BATCHN, 0, stream>>>(m, chk2edge);
        k_var<<<N_VARS, BATCHN, 0, stream>>>(m, llrT, outT, it == N_ITERS - 1);
        // outT [8192][512] -> out[it] [512][8192]
        int blocks = ((N_VARS / 16) * (BATCHN / 16)) / 16;   // 1024
        k_wmma_transpose<<<blocks, 512, 0, stream>>>(
            outT, out + (size_t)it * BATCHN * N_VARS, N_VARS, BATCHN);
    }
}